// DuelingDQN_24824910970960
// MI455X (gfx1250) — compile-verified
//
#include <hip/hip_runtime.h>
#include <math.h>

// ---------------------------------------------------------------------------
// DuelingDQN forward for MI455X (gfx1250, wave32).
//   B=8192, N_PIX=15000 (H=200, W=75), N_ACT=4.
// Phase A: permute W_lf into rgb-flat order (workspace).
// Phase B: bandwidth-bound masked reduction -> steering[B]  (HBM-limited, ~491MB).
//          rgb loads are non-temporal so the one-shot stream doesn't evict the
//          hot 60KB weight vector from L2 (192MB) between block re-stages.
// Phase C: dueling MLP via V_WMMA_F32_16X16X4_F32, 16 batch rows per wave.
// ---------------------------------------------------------------------------

typedef __attribute__((ext_vector_type(2))) float v2f;
typedef __attribute__((ext_vector_type(4))) float v4f;
typedef __attribute__((ext_vector_type(8))) float v8f;

#define NPIX   15000
#define NPIX4  3750      // NPIX / 4
#define BATCH  8192
#define ROWS_PER_BLOCK 4
#define HS     65        // LDS tile stride (odd -> bank-conflict friendly)

// ---------------------------------------------------------------------------
// Phase A: wperm[p] = W_lf[(p%200)*75 + p/200]
// ---------------------------------------------------------------------------
__global__ void permute_wlf_kernel(const float* __restrict__ W_lf,
                                   float* __restrict__ wperm) {
    int p = blockIdx.x * blockDim.x + threadIdx.x;
    if (p < NPIX) {
        int i = p % 200;   // row of the (200,75) mask
        int j = p / 200;   // col
        wperm[p] = W_lf[i * 75 + j];
    }
}

// ---------------------------------------------------------------------------
// Phase B: steering[r] = tanh( sum_p (rgb[r,p] > 0 ? wperm[p] : 0) + b_lf )
// One block handles 4 rows; permuted weights staged in LDS (60 KB) once.
// ---------------------------------------------------------------------------
__global__ __launch_bounds__(256)
void steering_kernel(const float* __restrict__ rgb,
                     const float* __restrict__ wperm,
                     const float* __restrict__ b_lf,
                     float* __restrict__ steer_out) {
    __shared__ v4f   wl4[NPIX4];      // 60000 bytes
    __shared__ float partial[8];

    const int tid = threadIdx.x;
    const v4f* wp4 = (const v4f*)wperm;
    for (int i = tid; i < NPIX4; i += 256) wl4[i] = wp4[i];
    __syncthreads();

    const float blf = b_lf[0];
    const int r0 = blockIdx.x * ROWS_PER_BLOCK;

    for (int rr = 0; rr < ROWS_PER_BLOCK; ++rr) {
        const int r = r0 + rr;
        const v4f* row = (const v4f*)(rgb + (size_t)r * NPIX);
        float acc = 0.f;
        for (int i = tid; i < NPIX4; i += 256) {
            // One-shot streaming read: non-temporal so it doesn't thrash L2.
            v4f v = __builtin_nontemporal_load(&row[i]);
            v4f w = wl4[i];
            acc += (v.x > 0.f ? w.x : 0.f) + (v.y > 0.f ? w.y : 0.f)
                 + (v.z > 0.f ? w.z : 0.f) + (v.w > 0.f ? w.w : 0.f);
        }
        // wave32 reduce
        #pragma unroll
        for (int off = 16; off > 0; off >>= 1)
            acc += __shfl_down(acc, off, 32);
        if ((tid & 31) == 0) partial[tid >> 5] = acc;
        __syncthreads();
        if (tid == 0) {
            float s = 0.f;
            #pragma unroll
            for (int w = 0; w < 8; ++w) s += partial[w];
            steer_out[r] = tanhf(s + blf);
        }
        __syncthreads();
    }
}

// ---------------------------------------------------------------------------
// Phase C: dueling MLP with V_WMMA_F32_16X16X4_F32.
// Fragment layout assumptions (per CDNA5 ISA §7.12.2):
//   A 16x4 f32 : lane L holds M = L%16;  VGPR v holds K = (L/16)*2 + v
//   B 4x16 f32 : lane L holds N = L%16;  VGPR v holds K = (L/16)*2 + v
//   C/D 16x16  : lane L holds N = L%16;  VGPR r holds M = r + (L/16)*8
// ---------------------------------------------------------------------------
template <int KIN, int NOUT, bool RELU>
__device__ __forceinline__
void dense_layer_wmma(const float* __restrict__ inb,   // LDS [16][HS]
                      float* __restrict__ outb,        // LDS [16][HS]
                      const float* __restrict__ W,     // global [KIN][NOUT]
                      const float* __restrict__ bias,  // global [NOUT]
                      int m, int hi) {
    #pragma unroll
    for (int nt = 0; nt < NOUT; nt += 16) {
        const int n = nt + m;  // this lane's output column
        v8f acc = {0.f, 0.f, 0.f, 0.f, 0.f, 0.f, 0.f, 0.f};
        #pragma unroll
        for (int k0 = 0; k0 < KIN; k0 += 4) {
            const int ka = k0 + hi * 2;
            v2f a, b;
            a.x = inb[m * HS + ka];
            a.y = inb[m * HS + ka + 1];
            b.x = W[ka * NOUT + n];
            b.y = W[(ka + 1) * NOUT + n];
            acc = __builtin_amdgcn_wmma_f32_16x16x4_f32(
                      false, a, false, b, (short)0, acc, false, false);
        }
        const float bn = bias[n];
        #pragma unroll
        for (int r = 0; r < 8; ++r) {
            float v = acc[r] + bn;
            if (RELU) v = fmaxf(v, 0.f);
            outb[(r + hi * 8) * HS + n] = v;
        }
    }
}

__global__ __launch_bounds__(128)
void mlp_wmma_kernel(const float* __restrict__ steer,
                     const float* __restrict__ dist,
                     const float* __restrict__ spd,
                     const float* __restrict__ W1, const float* __restrict__ b1,
                     const float* __restrict__ W2, const float* __restrict__ b2,
                     const float* __restrict__ W3, const float* __restrict__ b3,
                     const float* __restrict__ W4, const float* __restrict__ b4,
                     const float* __restrict__ Wv, const float* __restrict__ bv,
                     const float* __restrict__ Wa, const float* __restrict__ ba,
                     float* __restrict__ qout) {
    __shared__ float smem[4 * 2 * 16 * HS];   // 4 waves x 2 buffers x 16xHS

    const int tid  = threadIdx.x;
    const int lane = tid & 31;
    const int wid  = tid >> 5;
    const int m    = lane & 15;
    const int hi   = lane >> 4;

    float* bufA = smem + wid * (2 * 16 * HS);
    float* bufB = bufA + 16 * HS;

    const int tile = blockIdx.x * 4 + wid;   // 512 tiles total
    const int row0 = tile * 16;

    // ---- Layer 1: x[16,3] @ W1[3,32] (K padded 3->4 with zeros) -----------
    // All loads unconditional (always in-bounds); zero-padding via cndmask,
    // avoiding exec-mask branches.
    v2f ax;
    {
        const float s  = steer[row0 + m];
        const float d  = dist[row0 + m];
        const float sp = spd[row0 + m];
        ax.x = hi ? sp : s;      // K=2 : K=0
        ax.y = hi ? 0.f : d;     // K=3 (zero pad) : K=1
    }
    #pragma unroll
    for (int nt = 0; nt < 32; nt += 16) {
        const int n = nt + m;
        const float w1k0 = W1[(hi * 2) * 32 + n];   // K = 0 or 2 (both valid)
        const float w1k1 = W1[1 * 32 + n];          // row 1 (always valid)
        v2f bx;
        bx.x = w1k0;
        bx.y = hi ? 0.f : w1k1;                     // K = 3 -> zero pad
        v8f acc = {0.f, 0.f, 0.f, 0.f, 0.f, 0.f, 0.f, 0.f};
        acc = __builtin_amdgcn_wmma_f32_16x16x4_f32(
                  false, ax, false, bx, (short)0, acc, false, false);
        const float bn = b1[n];
        #pragma unroll
        for (int r = 0; r < 8; ++r)
            bufA[(r + hi * 8) * HS + n] = fmaxf(acc[r] + bn, 0.f);
    }
    __syncthreads();

    dense_layer_wmma<32, 64, true >(bufA, bufB, W2, b2, m, hi);
    __syncthreads();
    dense_layer_wmma<64, 64, true >(bufB, bufA, W3, b3, m, hi);
    __syncthreads();
    dense_layer_wmma<64, 32, false>(bufA, bufB, W4, b4, m, hi);  // no ReLU
    __syncthreads();

    // ---- Heads: value[1] + advantage[4], dueling combine (lanes 0..15) ----
    if (lane < 16) {
        const float* h = bufB + m * HS;
        float v  = bv[0];
        float a0 = ba[0], a1 = ba[1], a2 = ba[2], a3 = ba[3];
        #pragma unroll
        for (int k = 0; k < 32; ++k) {
            const float hv = h[k];
            v  = fmaf(hv, Wv[k],         v);
            a0 = fmaf(hv, Wa[k * 4 + 0], a0);
            a1 = fmaf(hv, Wa[k * 4 + 1], a1);
            a2 = fmaf(hv, Wa[k * 4 + 2], a2);
            a3 = fmaf(hv, Wa[k * 4 + 3], a3);
        }
        const float mean = 0.25f * (a0 + a1 + a2 + a3);
        const int ro = (row0 + m) * 4;
        qout[ro + 0] = v + a0 - mean;
        qout[ro + 1] = v + a1 - mean;
        qout[ro + 2] = v + a2 - mean;
        qout[ro + 3] = v + a3 - mean;
    }
}

// ---------------------------------------------------------------------------
extern "C" void kernel_launch(void* const* d_in, const int* in_sizes, int n_in,
                              void* d_out, int out_size, void* d_ws, size_t ws_size,
                              hipStream_t stream) {
    const float* rgb  = (const float*)d_in[0];
    const float* dist = (const float*)d_in[1];
    const float* spd  = (const float*)d_in[2];
    const float* W_lf = (const float*)d_in[3];
    const float* b_lf = (const float*)d_in[4];
    const float* W1 = (const float*)d_in[5];   const float* b1 = (const float*)d_in[6];
    const float* W2 = (const float*)d_in[7];   const float* b2 = (const float*)d_in[8];
    const float* W3 = (const float*)d_in[9];   const float* b3 = (const float*)d_in[10];
    const float* W4 = (const float*)d_in[11];  const float* b4 = (const float*)d_in[12];
    const float* Wv = (const float*)d_in[13];  const float* bv = (const float*)d_in[14];
    const float* Wa = (const float*)d_in[15];  const float* ba = (const float*)d_in[16];
    float* out = (float*)d_out;

    float* wperm = (float*)d_ws;          // [15000]
    float* steer = wperm + NPIX;          // [8192]

    // Phase A: permute lane-follower weights into rgb-flat order.
    permute_wlf_kernel<<<(NPIX + 255) / 256, 256, 0, stream>>>(W_lf, wperm);

    // Phase B: HBM-bound masked reduction -> steering.
    steering_kernel<<<BATCH / ROWS_PER_BLOCK, 256, 0, stream>>>(rgb, wperm, b_lf, steer);

    // Phase C: WMMA dueling MLP. 128 blocks x 4 waves x 16 rows = 8192 rows.
    mlp_wmma_kernel<<<BATCH / 64, 128, 0, stream>>>(
        steer, dist, spd,
        W1, b1, W2, b2, W3, b3, W4, b4, Wv, bv, Wa, ba, out);
}